// ChamferLoss_45260365365838
// MI455X (gfx1250) — compile-verified
//
#include <hip/hip_runtime.h>

typedef float v2f __attribute__((ext_vector_type(2)));
typedef float v8f __attribute__((ext_vector_type(8)));

#define B_SZ 16
#define NPTS 4096
#define WAVES 8
#define TPB (WAVES * 32)
#define ROWS_PER_BLOCK (WAVES * 16)                 // 128 rows per block
#define BLOCKS_PER_BATCH (NPTS / ROWS_PER_BLOCK)    // 32
#define NBLOCKS (B_SZ * BLOCKS_PER_BATCH)           // 512 blocks per pass

// One directed Chamfer pass: for each "row" point, min squared distance over
// all "col" points (same batch), via V_WMMA_F32_16X16X4_F32 Gram tiles.
//   rows: [B, NPTS, 3]  (points we take the argmin FOR)
//   cols: [B, NPTS, 3]  (points we minimize OVER; loop dimension)
//   idxOut: [B, NPTS]   argmin indices
//   partial: [NBLOCKS]  per-block sums of min squared distances
__global__ __launch_bounds__(TPB) void chamfer_dir_kernel(
    const float* __restrict__ rows,
    const float* __restrict__ cols,
    int* __restrict__ idxOut,
    float* __restrict__ partial)
{
    const int lane = threadIdx.x & 31;
    const int wave = threadIdx.x >> 5;
    const int b     = blockIdx.x / BLOCKS_PER_BATCH;
    const int chunk = blockIdx.x % BLOCKS_PER_BATCH;
    const int rowBase = chunk * ROWS_PER_BLOCK + wave * 16;
    const int half = lane >> 4;      // 0: lanes 0-15, 1: lanes 16-31
    const int l16  = lane & 15;

    // Each lane loads one of the wave's 16 row-points (lane and lane+16 share).
    const float* rp = rows + ((size_t)b * NPTS + rowBase + l16) * 3;
    const float px = rp[0], py = rp[1], pz = rp[2];
    const float p2self = px * px + py * py + pz * pz;

    // A operand (16x4 f32): lanes 0-15 carry (K0,K1)=(x,y); lanes 16-31 (K2,K3)=(z,0)
    v2f a;
    a.x = half ? pz : px;
    a.y = half ? 0.0f : py;

    // Squared norms of the 8 rows this lane's D registers cover:
    // D vgpr v <-> row (half*8 + v); that row's |P|^2 lives in lane (half*8 + v).
    float p2r[8];
#pragma unroll
    for (int v = 0; v < 8; ++v)
        p2r[v] = __shfl(p2self, half * 8 + v, 32);

    float minv[8];
    int   mini[8];
#pragma unroll
    for (int v = 0; v < 8; ++v) { minv[v] = 3.0e38f; mini[v] = 0; }

    const float* cbase = cols + (size_t)b * NPTS * 3;
    const int ntiles = NPTS / 16;

    for (int mt = 0; mt < ntiles; ++mt) {
        const int m = mt * 16 + l16;           // this lane's column index
        const float* gp = cbase + (size_t)m * 3;
        const float gx = gp[0], gy = gp[1], gz = gp[2];
        if (mt + 1 < ntiles)
            __builtin_prefetch(cbase + (size_t)(m + 16) * 3, 0, 0);
        const float g2 = gx * gx + gy * gy + gz * gz;

        // B operand (4x16 f32): same packing pattern as A
        v2f bb;
        bb.x = half ? gz : gx;
        bb.y = half ? 0.0f : gy;

        v8f acc = {};
        // D = A(16x4) x B(4x16): 16x16 tile of p.g dot products
        v8f pg = __builtin_amdgcn_wmma_f32_16x16x4_f32(
            false, a, false, bb, (short)0, acc, false, false);

#pragma unroll
        for (int v = 0; v < 8; ++v) {
            const float dist = p2r[v] + g2 - 2.0f * pg[v];
            if (dist < minv[v]) { minv[v] = dist; mini[v] = m; }
        }
    }

    // Min+argmin butterfly across the 16 lanes of each half (cols 0..15 mod 16).
#pragma unroll
    for (int v = 0; v < 8; ++v) {
#pragma unroll
        for (int mask = 8; mask >= 1; mask >>= 1) {
            const float ov = __shfl_xor(minv[v], mask, 32);
            const int   oi = __shfl_xor(mini[v], mask, 32);
            if (ov < minv[v] || (ov == minv[v] && oi < mini[v])) {
                minv[v] = ov; mini[v] = oi;
            }
        }
    }

    __shared__ float wsum[WAVES * 2];
    if (l16 == 0) {
        int* outp = idxOut + (size_t)b * NPTS + rowBase + half * 8;
        float localsum = 0.0f;
#pragma unroll
        for (int v = 0; v < 8; ++v) {
            outp[v] = mini[v];
            localsum += minv[v];
        }
        wsum[wave * 2 + half] = localsum;
    }
    __syncthreads();
    if (threadIdx.x == 0) {
        float s = 0.0f;
        for (int i = 0; i < WAVES * 2; ++i) s += wsum[i];
        partial[blockIdx.x] = s;
    }
}

// Deterministic fixed-order reduction of both passes' partial sums -> loss.
__global__ __launch_bounds__(256) void chamfer_finalize(
    const float* __restrict__ partial, float* __restrict__ out)
{
    __shared__ float sh[256];
    float s = 0.0f;
    for (int i = threadIdx.x; i < 2 * NBLOCKS; i += 256) s += partial[i];
    sh[threadIdx.x] = s;
    __syncthreads();
    for (int off = 128; off >= 1; off >>= 1) {
        if (threadIdx.x < off) sh[threadIdx.x] += sh[threadIdx.x + off];
        __syncthreads();
    }
    if (threadIdx.x == 0)
        out[0] = sh[0] / (float)(B_SZ * NPTS);   // N == M so one scale works
}

extern "C" void kernel_launch(void* const* d_in, const int* in_sizes, int n_in,
                              void* d_out, int out_size, void* d_ws, size_t ws_size,
                              hipStream_t stream) {
    const float* pred = (const float*)d_in[0];   // [B, N, 3] fp32
    const float* gt   = (const float*)d_in[1];   // [B, M, 3] fp32
    float* out  = (float*)d_out;                 // [0]=loss, then idx1, idx2 (int32 bits)
    int*   idx1 = (int*)(out + 1);
    int*   idx2 = (int*)(out + 1 + B_SZ * NPTS);
    float* partial = (float*)d_ws;               // 2*NBLOCKS floats of scratch

    // Pass 1: pred -> gt  (min over gt, writes idx1, partial[0:512))
    chamfer_dir_kernel<<<NBLOCKS, TPB, 0, stream>>>(pred, gt, idx1, partial);
    // Pass 2: gt -> pred  (min over pred, writes idx2, partial[512:1024))
    chamfer_dir_kernel<<<NBLOCKS, TPB, 0, stream>>>(gt, pred, idx2, partial + NBLOCKS);
    // Combine: loss = (sum pred2gt + sum gt2pred) / (B*N)
    chamfer_finalize<<<1, 256, 0, stream>>>(partial, out);
}